// SpectralConvLocal_66700842107120
// MI455X (gfx1250) — compile-verified
//
#include <hip/hip_runtime.h>

typedef float v2f __attribute__((ext_vector_type(2)));
typedef float v4f __attribute__((ext_vector_type(4)));
typedef float v8f __attribute__((ext_vector_type(8)));

#define CI_ 16
#define CO_ 16
#define KQ_ 8
#define NSTEP 68   /* 272 / 4 */

__global__ void zero_f32_kernel(float* __restrict__ p, int n) {
  int i = blockIdx.x * blockDim.x + threadIdx.x;
  if (i < n) p[i] = 0.0f;
}

// f [B, CI, N] -> fp [B, N, CI]
__global__ void transpose_f_kernel(const float* __restrict__ f,
                                   float* __restrict__ fp, int Bv, int Nv) {
  int idx = blockIdx.x * blockDim.x + threadIdx.x;
  int total = Bv * Nv * CI_;
  if (idx >= total) return;
  int i = idx % CI_;
  int n = (idx / CI_) % Nv;
  int b = idx / (CI_ * Nv);
  fp[idx] = f[((long long)b * CI_ + i) * Nv + n];
}

__global__ __launch_bounds__(256) void spectral_edges_kernel(
    const float* __restrict__ fsrc,    // fp [B,N,CI] if useTrans else f [B,CI,N]
    int useTrans,
    const float* __restrict__ bases_c, // [B,N,K]
    const float* __restrict__ bases_s, // [B,N,K]
    const int*   __restrict__ edges,   // [B,E,2]
    const float* __restrict__ nodew,   // [B,E]
    const float* __restrict__ w0,      // [CI,CO]
    const float* __restrict__ wc,      // [CI,CO,K]
    const float* __restrict__ wgs,     // [CI,CO,K]
    float* __restrict__ out,           // [B,CO,N]
    int Bv, int Nv, int Ev) {
  // Wbig[272][16] staged in WMMA-B lane layout: lw[(t*32+L)*2 + c] = Wbig[4t+2*(L>>4)+c][L&15]
  __shared__ float lw[NSTEP * 64];
  for (int idx = threadIdx.x; idx < NSTEP * 32; idx += blockDim.x) {
    int t = idx >> 5, L = idx & 31;
    int hh = L >> 4, col = L & 15;
    int j0 = 4 * t + 2 * hh;
#pragma unroll
    for (int c = 0; c < 2; ++c) {
      int j = j0 + c;
      float v;
      if (j < CI_) {
        v = w0[j * CO_ + col];
      } else if (j < CI_ + CI_ * KQ_) {
        int jj = j - CI_; int i = jj >> 3, k = jj & 7;
        v = 2.0f * wc[(i * CO_ + col) * KQ_ + k];
      } else {
        int jj = j - CI_ - CI_ * KQ_; int i = jj >> 3, k = jj & 7;
        v = 2.0f * wgs[(i * CO_ + col) * KQ_ + k];
      }
      lw[idx * 2 + c] = v;
    }
  }
  __syncthreads();

  const int lane = threadIdx.x & 31;
  const int h    = lane >> 4;    // lane half selects K-slice of A/B fragments
  const int le   = lane & 15;    // edge slot within 16-edge tile (also output channel)
  const long long totE  = (long long)Bv * Ev;
  const long long ntile = (totE + 15) >> 4;
  long long wid    = (long long)blockIdx.x * (blockDim.x >> 5) + (threadIdx.x >> 5);
  long long nwaves = (long long)gridDim.x * (blockDim.x >> 5);

  for (long long tile = wid; tile < ntile; tile += nwaves) {
    long long be = tile * 16 + le;
    bool valid = be < totE;
    long long beC = valid ? be : 0;
    int b = (int)(beC / Ev);
    int e = (int)(beC - (long long)b * Ev); (void)e;

    int tgt = edges[beC * 2 + 0];
    int src = edges[beC * 2 + 1];
    float nw = valid ? nodew[beC] : 0.0f;

    // gather + scale source feature (16 fp32)
    float fs[CI_];
    if (useTrans) {
      const float* row = fsrc + ((long long)b * Nv + src) * CI_;
#pragma unroll
      for (int q = 0; q < 4; ++q) {
        v4f v = *(const v4f*)(row + q * 4);
        fs[q * 4 + 0] = v.x * nw; fs[q * 4 + 1] = v.y * nw;
        fs[q * 4 + 2] = v.z * nw; fs[q * 4 + 3] = v.w * nw;
      }
    } else {
#pragma unroll
      for (int i = 0; i < CI_; ++i)
        fs[i] = fsrc[((long long)b * CI_ + i) * Nv + src] * nw;
    }

    // gather only the 4 k-values this lane half needs: k in {2h,2h+1,4+2h,5+2h}
    int k0 = 2 * h;
    const float* bcb = bases_c + (long long)b * Nv * KQ_;
    const float* bsb = bases_s + (long long)b * Nv * KQ_;
    v2f bct0 = *(const v2f*)(bcb + (long long)tgt * KQ_ + k0);
    v2f bct1 = *(const v2f*)(bcb + (long long)tgt * KQ_ + k0 + 4);
    v2f bcs0 = *(const v2f*)(bcb + (long long)src * KQ_ + k0);
    v2f bcs1 = *(const v2f*)(bcb + (long long)src * KQ_ + k0 + 4);
    v2f bst0 = *(const v2f*)(bsb + (long long)tgt * KQ_ + k0);
    v2f bst1 = *(const v2f*)(bsb + (long long)tgt * KQ_ + k0 + 4);
    v2f bss0 = *(const v2f*)(bsb + (long long)src * KQ_ + k0);
    v2f bss1 = *(const v2f*)(bsb + (long long)src * KQ_ + k0 + 4);

    float ccl[4], csl[4];
    ccl[0] = bct0.x * bcs0.x + bst0.x * bss0.x;
    ccl[1] = bct0.y * bcs0.y + bst0.y * bss0.y;
    ccl[2] = bct1.x * bcs1.x + bst1.x * bss1.x;
    ccl[3] = bct1.y * bcs1.y + bst1.y * bss1.y;
    csl[0] = bct0.x * bss0.x - bst0.x * bcs0.x;
    csl[1] = bct0.y * bss0.y - bst0.y * bcs0.y;
    csl[2] = bct1.x * bss1.x - bst1.x * bcs1.x;
    csl[3] = bct1.y * bss1.y - bst1.y * bcs1.y;

    // 68-step fp32 WMMA chain: msg[16x16] += A(16x4) * B(4x16)
    v8f acc = {};
#pragma unroll
    for (int t = 0; t < NSTEP; ++t) {
      v2f a;
      if (t < 4) {                       // identity section (w0): feat j = 4t+2h(+1)
        a.x = h ? fs[4 * t + 2] : fs[4 * t + 0];
        a.y = h ? fs[4 * t + 3] : fs[4 * t + 1];
      } else if (t < 36) {               // f (x) coef_c section
        int s = t - 4; int i = s >> 1; int kb = (s & 1) * 2;
        a.x = fs[i] * ccl[kb];
        a.y = fs[i] * ccl[kb + 1];
      } else {                           // f (x) coef_s section
        int s = t - 36; int i = s >> 1; int kb = (s & 1) * 2;
        a.x = fs[i] * csl[kb];
        a.y = fs[i] * csl[kb + 1];
      }
      v2f bfrag = *(const v2f*)(&lw[(t * 32 + lane) * 2]);
      acc = __builtin_amdgcn_wmma_f32_16x16x4_f32(
          false, a, false, bfrag, (short)0, acc, false, false);
    }

    // scatter-add: lane holds out-channel o=le for edges (tile*16 + r + 8h), r=0..7
    int o = le;
#pragma unroll
    for (int r = 0; r < 8; ++r) {
      int slot = r + 8 * h;
      long long be2 = tile * 16 + slot;
      int tg = __shfl(tgt, slot, 32);
      int b2 = __shfl(b, slot, 32);
      if (be2 < totE) {
        atomicAdd(&out[((long long)b2 * CO_ + o) * Nv + tg], acc[r]);
      }
    }
  }
}

extern "C" void kernel_launch(void* const* d_in, const int* in_sizes, int n_in,
                              void* d_out, int out_size, void* d_ws, size_t ws_size,
                              hipStream_t stream) {
  const float* f    = (const float*)d_in[0];
  const float* bc   = (const float*)d_in[1];
  const float* bs   = (const float*)d_in[2];
  /* d_in[3] = bases_0 : unused by the reference */
  const int*   edg  = (const int*)d_in[4];
  const float* nw   = (const float*)d_in[5];
  const float* wc   = (const float*)d_in[6];
  const float* wsn  = (const float*)d_in[7];
  const float* w0   = (const float*)d_in[8];

  const int Bv = 2;                                  // fixed by problem spec
  int M = in_sizes[8] / (CI_ * CO_);                 // == 1
  int N = (int)(in_sizes[3] / (long long)(M * Bv));  // bases_0 = B*N*1*M
  int E = (int)(in_sizes[5] / (long long)(M * Bv));  // node_weights = B*E*M

  float* outp = (float*)d_out;
  zero_f32_kernel<<<(out_size + 255) / 256, 256, 0, stream>>>(outp, out_size);

  size_t fpBytes = (size_t)Bv * N * CI_ * sizeof(float);
  int useTrans = (ws_size >= fpBytes) ? 1 : 0;
  const float* fsrc = f;
  if (useTrans) {
    int total = Bv * N * CI_;
    transpose_f_kernel<<<(total + 255) / 256, 256, 0, stream>>>(f, (float*)d_ws, Bv, N);
    fsrc = (const float*)d_ws;
  }

  spectral_edges_kernel<<<4096, 256, 0, stream>>>(
      fsrc, useTrans, bc, bs, edg, nw, w0, wc, wsn, outp, Bv, N, E);
}